// GCN_modified_5772436045962
// MI455X (gfx1250) — compile-verified
//
#include <hip/hip_runtime.h>

typedef __attribute__((ext_vector_type(2))) float v2f;
typedef __attribute__((ext_vector_type(8))) float v8f;

// ---------------------------------------------------------------------------
// Dense GEMM  C[M,N] = A[M,K] @ B[K,N]  via V_WMMA_F32_16X16X4_F32.
// One wave per 16x16 output tile. A-layout: lane = (g*16 + r), r=M, VGPR v
// holds K = k0 + g*2 + v. B-layout mirrored: r=N, VGPR v holds K = k0+g*2+v.
// C/D layout: VGPR i -> M = g*8 + i, N = r.
// ---------------------------------------------------------------------------
__global__ __launch_bounds__(256) void gcn_gemm_wmma_f32(
    const float* __restrict__ A, const float* __restrict__ B,
    float* __restrict__ C, int M, int N, int K) {
  const int wavesPerBlock = blockDim.x >> 5;
  const int wave = blockIdx.x * wavesPerBlock + (threadIdx.x >> 5);
  const int tilesN = N >> 4;
  const int tileM = wave / tilesN;
  const int tileN = wave % tilesN;
  if (tileM * 16 >= M) return;          // uniform per wave
  const int lane = threadIdx.x & 31;
  const int g = lane >> 4;              // half-wave select
  const int r = lane & 15;              // row (A) / col (B,C)

  v8f acc = {};
  const float* Arow = A + (size_t)(tileM * 16 + r) * K;
  const float* Bcol = B + (size_t)(tileN * 16 + r);

  for (int k0 = 0; k0 < K; k0 += 4) {
    const int ka = k0 + g * 2;
    v2f a; a[0] = Arow[ka];                 a[1] = Arow[ka + 1];
    v2f b; b[0] = Bcol[(size_t)ka * N];     b[1] = Bcol[(size_t)(ka + 1) * N];
    acc = __builtin_amdgcn_wmma_f32_16x16x4_f32(
        /*neg_a=*/false, a, /*neg_b=*/false, b,
        /*c_mod=*/(short)0, acc, /*reuse_a=*/false, /*reuse_b=*/false);
  }

  float* Cp = C + (size_t)(tileM * 16 + g * 8) * N + tileN * 16 + r;
#pragma unroll
  for (int i = 0; i < 8; ++i) Cp[(size_t)i * N] = acc[i];
}

// ---------------------------------------------------------------------------
// SpMM scatter:  acc[rows[e], :] += vals[e] * dense[cols[e], :]
// One thread per (edge, 4-float chunk). dense (<=102.4MB) is L2-resident, so
// the random gather + f32 atomic scatter stay inside the 192MB L2.
// ---------------------------------------------------------------------------
__global__ __launch_bounds__(256) void gcn_spmm_scatter(
    const int* __restrict__ rows, const int* __restrict__ cols,
    const float* __restrict__ vals, const float* __restrict__ dense,
    float* __restrict__ acc, int D, int nE) {
  const long long tid = (long long)blockIdx.x * blockDim.x + threadIdx.x;
  const int chunks = D >> 2;
  const int e = (int)(tid / chunks);
  if (e >= nE) return;
  const int c = ((int)(tid % chunks)) << 2;

  const float v = vals[e];
  const float4 d = *(const float4*)(dense + (size_t)cols[e] * D + c);
  float* dst = acc + (size_t)rows[e] * D + c;
  atomicAdd(dst + 0, v * d.x);
  atomicAdd(dst + 1, v * d.y);
  atomicAdd(dst + 2, v * d.z);
  atomicAdd(dst + 3, v * d.w);
}

// ---------------------------------------------------------------------------
// h = relu(acc + b)   (D is a power of two: 256)
// ---------------------------------------------------------------------------
__global__ __launch_bounds__(256) void gcn_bias_relu(
    const float* __restrict__ acc, const float* __restrict__ bias,
    float* __restrict__ out, int D, long long total) {
  const long long i = (long long)blockIdx.x * blockDim.x + threadIdx.x;
  if (i >= total) return;
  const float xv = acc[i] + bias[(int)(i & (long long)(D - 1))];
  out[i] = xv > 0.f ? xv : 0.f;
}

// ---------------------------------------------------------------------------
// out = log_softmax(acc + b) over rows of 64; one wave32 per row, 2 elems/lane.
// ---------------------------------------------------------------------------
__global__ __launch_bounds__(256) void gcn_bias_logsoftmax(
    const float* __restrict__ acc, const float* __restrict__ bias,
    float* __restrict__ out, int nRows) {
  const int wave = blockIdx.x * (blockDim.x >> 5) + (threadIdx.x >> 5);
  if (wave >= nRows) return;
  const int lane = threadIdx.x & 31;

  const float* in = acc + (size_t)wave * 64;
  float x0 = in[lane] + bias[lane];
  float x1 = in[lane + 32] + bias[lane + 32];

  float m = fmaxf(x0, x1);
#pragma unroll
  for (int off = 16; off > 0; off >>= 1) m = fmaxf(m, __shfl_xor(m, off, 32));

  float s = expf(x0 - m) + expf(x1 - m);
#pragma unroll
  for (int off = 16; off > 0; off >>= 1) s += __shfl_xor(s, off, 32);

  const float lse = m + logf(s);
  float* o = out + (size_t)wave * 64;
  o[lane] = x0 - lse;
  o[lane + 32] = x1 - lse;
}

// ---------------------------------------------------------------------------
// Orchestration (all on `stream`, graph-capture safe).
// ---------------------------------------------------------------------------
extern "C" void kernel_launch(void* const* d_in, const int* in_sizes, int n_in,
                              void* d_out, int out_size, void* d_ws, size_t ws_size,
                              hipStream_t stream) {
  const float* x   = (const float*)d_in[0];
  const int*   a1r = (const int*)  d_in[1];
  const int*   a1c = (const int*)  d_in[2];
  const float* a1v = (const float*)d_in[3];
  const int*   a2r = (const int*)  d_in[4];
  const int*   a2c = (const int*)  d_in[5];
  const float* a2v = (const float*)d_in[6];
  const float* W1  = (const float*)d_in[7];
  const float* b1  = (const float*)d_in[8];
  const float* W2  = (const float*)d_in[9];
  const float* b2  = (const float*)d_in[10];
  float* out = (float*)d_out;

  constexpr int Nn = 100000, Ee = 3200000, NF = 512, NH = 256, NC = 64;

  // workspace layout: s1[Nn*NH] | acc1[Nn*NH]; acc1 region later reused for
  // support2 [Nn*NC] (offset 0) and acc2 [Nn*NC] (offset Nn*NC).
  float* s1   = (float*)d_ws;
  float* acc1 = s1 + (size_t)Nn * NH;
  float* s2   = acc1;
  float* acc2 = acc1 + (size_t)Nn * NC;

  // Layer 1 -----------------------------------------------------------------
  hipMemsetAsync(acc1, 0, (size_t)Nn * NH * sizeof(float), stream);

  {   // support1 = x @ W1
    const int waves = (Nn / 16) * (NH / 16);               // 100000
    gcn_gemm_wmma_f32<<<waves / 8, 256, 0, stream>>>(x, W1, s1, Nn, NH, NF);
  }
  {   // acc1 += adj1 @ support1
    const long long threads = (long long)Ee * (NH / 4);    // 204.8M
    gcn_spmm_scatter<<<(int)(threads / 256), 256, 0, stream>>>(
        a1r, a1c, a1v, s1, acc1, NH, Ee);
  }
  {   // h = relu(acc1 + b1)  -> overwrite s1
    const long long total = (long long)Nn * NH;
    gcn_bias_relu<<<(int)((total + 255) / 256), 256, 0, stream>>>(
        acc1, b1, s1, NH, total);
  }

  // Layer 2 -----------------------------------------------------------------
  hipMemsetAsync(acc2, 0, (size_t)Nn * NC * sizeof(float), stream);

  {   // support2 = h @ W2
    const int waves = (Nn / 16) * (NC / 16);               // 25000
    gcn_gemm_wmma_f32<<<waves / 8, 256, 0, stream>>>(s1, W2, s2, Nn, NC, NH);
  }
  {   // acc2 += adj2 @ support2
    const long long threads = (long long)Ee * (NC / 4);    // 51.2M
    gcn_spmm_scatter<<<(int)(threads / 256), 256, 0, stream>>>(
        a2r, a2c, a2v, s2, acc2, NC, Ee);
  }
  {   // out = log_softmax(acc2 + b2)
    gcn_bias_logsoftmax<<<(Nn + 7) / 8, 256, 0, stream>>>(acc2, b2, out, Nn);
  }
}